// SparseEquivariantLayerBlock_18425409699998
// MI455X (gfx1250) — compile-verified
//
#include <hip/hip_runtime.h>

typedef float v2f __attribute__((ext_vector_type(2)));
typedef float v8f __attribute__((ext_vector_type(8)));

#define NN    256
#define CIN   96
#define COUT  96
#define BATCH 4
#define PLANE (NN * NN)   // 65536

// ---------------------------------------------------------------------------
// Kernel 1: per-(b,c) plane reductions: diag, row sums, col sums, trace, sum.
// One block per (b,c). Thread t owns column q=t (coalesced reads across block).
// ---------------------------------------------------------------------------
__global__ __launch_bounds__(256) void sel_stats_kernel(
    const float* __restrict__ X,
    float* __restrict__ dg, float* __restrict__ rs, float* __restrict__ cs,
    float* __restrict__ tr, float* __restrict__ sm)
{
    __shared__ float rowpart[NN * 8];
    __shared__ float red[16];
    const int bc   = blockIdx.x;          // b*CIN + c
    const int tid  = threadIdx.x;         // 0..255 (column index)
    const int wave = tid >> 5;
    const int lane = tid & 31;
    const float* Xp = X + (size_t)bc * PLANE;

    float colsum = 0.f, dval = 0.f;
    for (int p = 0; p < NN; ++p) {
        float v = Xp[p * NN + tid];
        colsum += v;
        if (p == tid) dval = v;
        float part = v;
        #pragma unroll
        for (int off = 16; off >= 1; off >>= 1)
            part += __shfl_xor(part, off, 32);
        if (lane == 0) rowpart[p * 8 + wave] = part;
    }
    __syncthreads();

    float rsum = 0.f;
    #pragma unroll
    for (int w = 0; w < 8; ++w) rsum += rowpart[tid * 8 + w];
    rs[bc * NN + tid] = rsum;
    cs[bc * NN + tid] = colsum;
    dg[bc * NN + tid] = dval;

    // trace and total sum
    float tpart = dval, spart = colsum;
    #pragma unroll
    for (int off = 16; off >= 1; off >>= 1) {
        tpart += __shfl_xor(tpart, off, 32);
        spart += __shfl_xor(spart, off, 32);
    }
    if (lane == 0) { red[wave] = tpart; red[8 + wave] = spart; }
    __syncthreads();
    if (tid == 0) {
        float tt = 0.f, ss = 0.f;
        #pragma unroll
        for (int w = 0; w < 8; ++w) { tt += red[w]; ss += red[8 + w]; }
        tr[bc] = tt;
        sm[bc] = ss;
    }
}

// ---------------------------------------------------------------------------
// Kernel 2: channel-mix the small stats into per-(b,o) epilogue vectors.
//   rowvec[p]  = sum_c (w5*d + w7*c + w12*r)
//   colvec[q]  = sum_c (w9*d + w10*c + w13*r)
//   diagvec[p] = sum_c (w0*d + w1*c + w3*r + w2*t + w4*S)
//   scal       = sum_c (w11*t + w14*S) + bias
// Weight-index mapping derived from PARTITIONS enumeration order.
// ---------------------------------------------------------------------------
__global__ __launch_bounds__(256) void sel_mix_kernel(
    const float* __restrict__ W, const float* __restrict__ bias,
    const float* __restrict__ dg, const float* __restrict__ rs,
    const float* __restrict__ cs, const float* __restrict__ tr,
    const float* __restrict__ sm,
    float* __restrict__ rowvec, float* __restrict__ colvec,
    float* __restrict__ diagvec, float* __restrict__ scal)
{
    const int o = blockIdx.x;     // output channel (uniform -> s_loads of W)
    const int b = blockIdx.y;
    const int p = threadIdx.x;

    float rv = 0.f, cv = 0.f, dv = 0.f, sc = 0.f;
    for (int c = 0; c < CIN; ++c) {
        const int   bcn = (b * CIN + c) * NN;
        const float dd = dg[bcn + p];
        const float rr = rs[bcn + p];
        const float cc = cs[bcn + p];
        const float tt = tr[b * CIN + c];
        const float ss = sm[b * CIN + c];
        #define WGT(i) W[((i) * CIN + c) * COUT + o]
        dv += WGT(0) * dd + WGT(1)  * cc + WGT(3)  * rr + WGT(2) * tt + WGT(4) * ss;
        rv += WGT(5) * dd + WGT(7)  * cc + WGT(12) * rr;
        cv += WGT(9) * dd + WGT(10) * cc + WGT(13) * rr;
        sc += WGT(11) * tt + WGT(14) * ss;
        #undef WGT
    }
    const int bo = (b * COUT + o);
    rowvec [bo * NN + p] = rv;
    colvec [bo * NN + p] = cv;
    diagvec[bo * NN + p] = dv;
    if (p == 0) scal[bo] = sc + bias[0];
}

// ---------------------------------------------------------------------------
// Kernel 3: main fused GEMM + epilogue.
// Block = one 16x16 spatial tile (p0,q0) of one batch. 12 waves:
//   orient 0 (waves 0..5):  o-tile = wave, D = W8^T * X[:, p0+mg, q0+ln]
//   orient 1 (waves 6..11): o-tile = wave-6, D = W6^T * X[:, q0+mg, p0+ln]
// Both accumulate into LDS Yt[o][pi][qi] via ds_add_f32, then vectorized
// epilogue adds rowvec/colvec/diag/scalar and stores Y.
// WMMA: V_WMMA_F32_16X16X4_F32, K chained over 96 channels (24 steps).
// ---------------------------------------------------------------------------
__global__ __launch_bounds__(384) void sel_gemm_kernel(
    const float* __restrict__ X, const float* __restrict__ W,
    const float* __restrict__ rowvec, const float* __restrict__ colvec,
    const float* __restrict__ diagvec, const float* __restrict__ scal,
    float* __restrict__ Y)
{
    __shared__ float Yt[COUT * 256];   // 96 KB: [o][pi][qi]

    const int tid    = threadIdx.x;
    const int wave   = tid >> 5;
    const int lane   = tid & 31;
    const int hi     = lane >> 4;      // lane half (selects K pair / M+8)
    const int ln     = lane & 15;
    const int orient = wave / 6;       // 0: identity term (w8), 1: transpose term (w6)
    const int ot     = wave % 6;       // o-tile
    const int tile   = blockIdx.x;
    const int b      = blockIdx.y;
    const int p0     = (tile >> 4) << 4;
    const int q0     = (tile & 15) << 4;

    // zero the LDS accumulation tile
    for (int i = tid; i < COUT * 256; i += 384) Yt[i] = 0.f;

    // Preload A fragments: A[o_row=ln, k] ; VGPR0 holds K=k0+2*hi, VGPR1 K=k0+1+2*hi
    const int widx = orient ? 6 : 8;   // transpose weight : identity weight
    v2f afrag[24];
    #pragma unroll
    for (int k = 0; k < 24; ++k) {
        const int c0 = 4 * k + 2 * hi;
        afrag[k].x = W[(widx * CIN + c0    ) * COUT + ot * 16 + ln];
        afrag[k].y = W[(widx * CIN + c0 + 1) * COUT + ot * 16 + ln];
    }
    __syncthreads();

    const int row0 = orient ? q0 : p0;
    const int col0 = orient ? p0 : q0;
    const float* Xb = X + (size_t)b * CIN * PLANE;

    for (int mg = 0; mg < 16; ++mg) {
        // B fragment base: lane ln of half 'hi' reads channels (4k+2hi, 4k+1+2hi)
        const float* xp = Xb + (size_t)(2 * hi) * PLANE
                             + (size_t)(row0 + mg) * NN + col0 + ln;
        v8f acc = {0.f, 0.f, 0.f, 0.f, 0.f, 0.f, 0.f, 0.f};
        #pragma unroll
        for (int k = 0; k < 24; ++k) {
            v2f bf;
            bf.x = xp[0];
            bf.y = xp[PLANE];
            acc = __builtin_amdgcn_wmma_f32_16x16x4_f32(
                    false, afrag[k], false, bf, (short)0, acc, false, false);
            xp += 4 * (size_t)PLANE;
        }
        // D layout: VGPR j -> (o_local = j + 8*hi, n = ln)
        const int pi = orient ? ln : mg;
        const int qi = orient ? mg : ln;
        #pragma unroll
        for (int j = 0; j < 8; ++j) {
            const int o = ot * 16 + j + 8 * hi;
            atomicAdd(&Yt[o * 256 + pi * 16 + qi], acc[j]);   // ds_add_f32
        }
    }
    __syncthreads();

    // Epilogue: Yt + rowvec(p) + colvec(q) + scal + diag(p==q), float4 stores
    for (int i4 = tid; i4 < COUT * 64; i4 += 384) {
        const int o  = i4 >> 6;
        const int pi = (i4 >> 2) & 15;
        const int q4 = (i4 & 3) * 4;
        const int bo = b * COUT + o;
        float4 v = *(const float4*)&Yt[i4 * 4];
        const float  base = rowvec[bo * NN + p0 + pi] + scal[bo];
        const float4 cv   = *(const float4*)&colvec[bo * NN + q0 + q4];
        v.x += base + cv.x; v.y += base + cv.y;
        v.z += base + cv.z; v.w += base + cv.w;
        const int p = p0 + pi, qb = q0 + q4;
        if (p >= qb && p < qb + 4) {
            const float dgv = diagvec[bo * NN + p];
            const int j = p - qb;
            if      (j == 0) v.x += dgv;
            else if (j == 1) v.y += dgv;
            else if (j == 2) v.z += dgv;
            else             v.w += dgv;
        }
        *(float4*)&Y[(size_t)bo * PLANE + (size_t)p * NN + qb] = v;
    }
}

// ---------------------------------------------------------------------------
extern "C" void kernel_launch(void* const* d_in, const int* in_sizes, int n_in,
                              void* d_out, int out_size, void* d_ws, size_t ws_size,
                              hipStream_t stream)
{
    const float* X    = (const float*)d_in[0];
    // d_in[1] = Y_in (zeros, unused by the math)
    const float* W    = (const float*)d_in[2];
    const float* bias = (const float*)d_in[3];

    float* ws      = (float*)d_ws;
    float* dg      = ws;                    // [4][96][256]
    float* rs      = dg      + BATCH * CIN * NN;
    float* cs      = rs      + BATCH * CIN * NN;
    float* tr      = cs      + BATCH * CIN * NN;   // [4][96]
    float* sm      = tr      + BATCH * CIN;
    float* rowvec  = sm      + BATCH * CIN;        // [4][96][256]
    float* colvec  = rowvec  + BATCH * COUT * NN;
    float* diagvec = colvec  + BATCH * COUT * NN;
    float* scal    = diagvec + BATCH * COUT * NN;  // [4][96]
    float* Y       = (float*)d_out;

    sel_stats_kernel<<<dim3(BATCH * CIN), dim3(256), 0, stream>>>(
        X, dg, rs, cs, tr, sm);

    sel_mix_kernel<<<dim3(COUT, BATCH), dim3(256), 0, stream>>>(
        W, bias, dg, rs, cs, tr, sm, rowvec, colvec, diagvec, scal);

    sel_gemm_kernel<<<dim3(256, BATCH), dim3(384), 0, stream>>>(
        X, W, rowvec, colvec, diagvec, scal, Y);
}